// MorphAudioModel_15161234555500
// MI455X (gfx1250) — compile-verified
//
#include <hip/hip_runtime.h>
#include <hip/hip_bf16.h>
#include <math.h>

typedef __attribute__((ext_vector_type(2))) float v2f;
typedef __attribute__((ext_vector_type(8))) float v8f;

#define N1 15996     // conv1 'valid' output length (batch0, ch0 only is live)
#define N2 5328      // conv2 'valid' output length
#define NF 5584      // live prefix of flattened pool2 output (max idx 5583)
#define KFC 7104     // FC1 K  (= 4*1776)
#define KT 192       // K tile (7104 = 37*192)
#define NKT 37
#define BSTRIDE 193  // 193 % 64 == 1 -> conflict-free LDS rows
#define LDS_B_OFF 5600

// ---------- stage 1: conv1 (batch0, out-channel0 only) + relu ----------
__global__ void k_conv1(const float* __restrict__ x, const float* __restrict__ w,
                        const float* __restrict__ b, float* __restrict__ y1) {
    int m = blockIdx.x * blockDim.x + threadIdx.x;
    if (m >= N1) return;
    float acc = b[0];
#pragma unroll
    for (int k = 0; k < 5; ++k) acc += x[m + k] * w[k];   // cross-correlation, like torch/jax
    y1[m] = fmaxf(acc, 0.0f);
}

// ---------- stage 2: parabolic max-plus pool (full, unstrided), channel0 ----------
__global__ void k_pool1(const float* __restrict__ y1, const float* __restrict__ t1,
                        float* __restrict__ p1) {
    int m = blockIdx.x * blockDim.x + threadIdx.x;
    if (m >= N1) return;
    float inv4t = 1.0f / (4.0f * t1[0]);
    float best = -INFINITY;
#pragma unroll
    for (int d = 0; d < 7; ++d) {
        int i = m + d - 3;
        if (i >= 0 && i < N1) {
            float z = (float)(d - 3);
            best = fmaxf(best, y1[i] - z * z * inv4t);
        }
    }
    p1[m] = best;
}

// ---------- stage 3: conv2 + relu, only out-channels {0,1} are live ----------
// aliased input: in[ic][n] = p1[3n + ic]
__global__ void k_conv2(const float* __restrict__ p1, const float* __restrict__ w2,
                        const float* __restrict__ b2, float* __restrict__ y2) {
    int idx = blockIdx.x * blockDim.x + threadIdx.x;
    if (idx >= 2 * N2) return;
    int c = idx / N2, n = idx % N2;
    float acc = b2[c];
#pragma unroll
    for (int ic = 0; ic < 2; ++ic)
#pragma unroll
        for (int k = 0; k < 5; ++k)
            acc += p1[3 * (n + k) + ic] * w2[c * 10 + ic * 5 + k];
    y2[c * N2 + n] = fmaxf(acc, 0.0f);
}

// ---------- stage 4: pool2 -> live prefix of flattened buffer (flat2[0..5583]) ----------
__global__ void k_pool2(const float* __restrict__ y2, const float* __restrict__ t2,
                        float* __restrict__ flat2) {
    int i = blockIdx.x * blockDim.x + threadIdx.x;
    if (i >= NF) return;
    int c = (i < N2) ? 0 : 1;
    int m = i - c * N2;
    float inv4t = 1.0f / (4.0f * t2[c]);
    float best = -INFINITY;
#pragma unroll
    for (int d = 0; d < 7; ++d) {
        int j = m + d - 3;
        if (j >= 0 && j < N2) {
            float z = (float)(d - 3);
            best = fmaxf(best, y2[c * N2 + j] - z * z * inv4t);
        }
    }
    flat2[i] = best;
}

// ---------- stage 5: FC1 as WMMA f32 GEMM ----------
// D[b,o] = sum_k feat[b,k] * W[o,k],  feat[b,k] = flat2[b + (k/1776) + 3*(k%1776)]
// A frag (16x4 f32): lane L holds A[M = L&15, K = q + 2*(L>>4)], q = vgpr 0/1
// B frag (4x16 f32): lane L holds B[K = q + 2*(L>>4), N = L&15]
// C/D (16x16 f32):  vgpr v, lane L -> D[v + 8*(L>>4), L&15]
__global__ __launch_bounds__(32) void k_fc1_wmma(const float* __restrict__ flat2,
                                                 const float* __restrict__ w,
                                                 const float* __restrict__ bias,
                                                 float* __restrict__ out1) {
    __shared__ float lds[LDS_B_OFF + 16 * BSTRIDE];   // flat2 (5584->5600) + B tile
    const int lane = threadIdx.x;
    const int mt = blockIdx.x;        // 0..15  (batch rows r0..r0+15)
    const int nt = blockIdx.y;        // 0..6   (neuron cols o0..o0+15, padded to 112)
    const int r0 = mt * 16, o0 = nt * 16;
    const int hi = lane >> 4, lo = lane & 15;

    // whole live feature vector into LDS once (22.4 KB)
    for (int i = lane; i < NF; i += 32) lds[i] = flat2[i];

    v8f acc = {};
    for (int kt = 0; kt < NKT; ++kt) {
        const int kbase = kt * KT;
        // stage 16 x 192 weight tile (rows >= 100 zero-padded), coalesced
        for (int i = lane; i < 16 * KT; i += 32) {
            int row = i / KT, col = i - row * KT;
            int o = o0 + row;
            lds[LDS_B_OFF + row * BSTRIDE + col] =
                (o < 100) ? w[o * KFC + kbase + col] : 0.0f;
        }
        __syncthreads();   // single wave: S_NOP, DS ops are in-order anyway
#pragma unroll 4
        for (int s = 0; s < KT / 4; ++s) {
            int kl = s * 4 + 2 * hi;          // local K of vgpr q=0
            int kg0 = kbase + kl;
            int kg1 = kg0 + 1;
            // feature gather via the as_strided aliasing: idx = row + c + 3n
            unsigned c0 = (unsigned)kg0 / 1776u, n0 = (unsigned)kg0 - c0 * 1776u;
            unsigned c1 = (unsigned)kg1 / 1776u, n1 = (unsigned)kg1 - c1 * 1776u;
            v2f a, b;
            a.x = lds[r0 + lo + (int)c0 + 3 * (int)n0];
            a.y = lds[r0 + lo + (int)c1 + 3 * (int)n1];
            b.x = lds[LDS_B_OFF + lo * BSTRIDE + kl];
            b.y = lds[LDS_B_OFF + lo * BSTRIDE + kl + 1];
            acc = __builtin_amdgcn_wmma_f32_16x16x4_f32(
                false, a, false, b, (short)0, acc, false, false);
        }
        __syncthreads();
    }
    // bias + relu + store (skip padded columns >= 100)
    const int col = o0 + lo;
    if (col < 100) {
#pragma unroll
        for (int v = 0; v < 8; ++v) {
            int row = r0 + v + 8 * hi;
            out1[row * 100 + col] = fmaxf(acc[v] + bias[col], 0.0f);
        }
    }
}

// ---------- stage 6: FC2 + log_softmax ----------
__global__ void k_fc2(const float* __restrict__ out1, const float* __restrict__ w,
                      const float* __restrict__ b, float* __restrict__ out) {
    int batch = blockIdx.x * blockDim.x + threadIdx.x;
    if (batch >= 256) return;
    float z[10];
#pragma unroll
    for (int k = 0; k < 10; ++k) z[k] = b[k];
    for (int o = 0; o < 100; ++o) {
        float f = out1[batch * 100 + o];
#pragma unroll
        for (int k = 0; k < 10; ++k) z[k] += f * w[k * 100 + o];
    }
    float m = z[0];
#pragma unroll
    for (int k = 1; k < 10; ++k) m = fmaxf(m, z[k]);
    float s = 0.0f;
#pragma unroll
    for (int k = 0; k < 10; ++k) s += expf(z[k] - m);
    float lse = logf(s) + m;
#pragma unroll
    for (int k = 0; k < 10; ++k) out[batch * 10 + k] = z[k] - lse;
}

extern "C" void kernel_launch(void* const* d_in, const int* in_sizes, int n_in,
                              void* d_out, int out_size, void* d_ws, size_t ws_size,
                              hipStream_t stream) {
    const float* x       = (const float*)d_in[0];   // (256,1,16000)
    const float* conv1_w = (const float*)d_in[1];   // (2,1,5)
    const float* conv1_b = (const float*)d_in[2];   // (2,)
    const float* t1      = (const float*)d_in[3];   // (3,)
    const float* conv2_w = (const float*)d_in[4];   // (4,2,5)
    const float* conv2_b = (const float*)d_in[5];   // (4,)
    const float* t2      = (const float*)d_in[6];   // (5,)
    const float* fc1_w   = (const float*)d_in[7];   // (100,7104)
    const float* fc1_b   = (const float*)d_in[8];   // (100,)
    const float* fc2_w   = (const float*)d_in[9];   // (10,100)
    const float* fc2_b   = (const float*)d_in[10];  // (10,)
    float* out = (float*)d_out;                     // (256,10)

    float* Y1    = (float*)d_ws;       // 16000
    float* P1    = Y1 + 16000;         // 16000
    float* Y2    = P1 + 16000;         // 10688 (uses 2*5328)
    float* FLAT2 = Y2 + 10688;         // 5600  (uses 5584)
    float* OUT1  = FLAT2 + 5600;       // 25600 (256*100)
    (void)in_sizes; (void)n_in; (void)out_size; (void)ws_size;

    k_conv1<<<(N1 + 255) / 256, 256, 0, stream>>>(x, conv1_w, conv1_b, Y1);
    k_pool1<<<(N1 + 255) / 256, 256, 0, stream>>>(Y1, t1, P1);
    k_conv2<<<(2 * N2 + 255) / 256, 256, 0, stream>>>(P1, conv2_w, conv2_b, Y2);
    k_pool2<<<(NF + 255) / 256, 256, 0, stream>>>(Y2, t2, FLAT2);
    k_fc1_wmma<<<dim3(16, 7), 32, 0, stream>>>(FLAT2, fc1_w, fc1_b, OUT1);
    k_fc2<<<8, 32, 0, stream>>>(OUT1, fc2_w, fc2_b, out);
}